// MultiSpeciesResistanceModel_35837207118200
// MI455X (gfx1250) — compile-verified
//
#include <hip/hip_runtime.h>
#include <hip/hip_bf16.h>
#include <math.h>

#define NNODES   4096
#define NEDGES   131072
#define MAT_ELEMS (NNODES * NNODES)          // 16777216
#define NS_ITERS 40                           // even -> final X lands back in ws buffer

// ---- GEMM tiling ----
#define BM 128
#define BN 128
#define BK 16
#define LDA_S 20    // A-tile LDS row stride: 80B (16B aligned); frag reads hit disjoint banks
#define LDB_S 136   // B-tile LDS row stride: 2*136 % 64 == 16 -> half-wave bank ranges disjoint

typedef __attribute__((ext_vector_type(2))) float v2f;
typedef __attribute__((ext_vector_type(4))) float v4f;
typedef __attribute__((ext_vector_type(8))) float v8f;

// ---------------------------------------------------------------------------
// Edge MLPs: shared + species heads, conductance = 1/(softplus(s+p)+1e-4)
// ---------------------------------------------------------------------------
__device__ __forceinline__ float mlp16_32_32_1(const float* x,
                                               const float* __restrict__ W1, const float* __restrict__ b1,
                                               const float* __restrict__ W2, const float* __restrict__ b2,
                                               const float* __restrict__ W3, const float* __restrict__ b3) {
    float h1[32];
    #pragma unroll
    for (int h = 0; h < 32; ++h) {
        float a = b1[h];
        #pragma unroll
        for (int k = 0; k < 16; ++k) a = fmaf(x[k], W1[k * 32 + h], a);
        h1[h] = fmaxf(a, 0.0f);
    }
    float h2[32];
    #pragma unroll 4
    for (int h = 0; h < 32; ++h) {
        float a = b2[h];
        #pragma unroll
        for (int k = 0; k < 32; ++k) a = fmaf(h1[k], W2[k * 32 + h], a);
        h2[h] = fmaxf(a, 0.0f);
    }
    float out = b3[0];
    #pragma unroll
    for (int k = 0; k < 32; ++k) out = fmaf(h2[k], W3[k], out);
    return out;
}

__global__ __launch_bounds__(256) void mlp_edges_kernel(
    const float* __restrict__ ef,
    const float* __restrict__ sW1, const float* __restrict__ sb1,
    const float* __restrict__ sW2, const float* __restrict__ sb2,
    const float* __restrict__ sW3, const float* __restrict__ sb3,
    const float* __restrict__ pW1, const float* __restrict__ pb1,
    const float* __restrict__ pW2, const float* __restrict__ pb2,
    const float* __restrict__ pW3, const float* __restrict__ pb3,
    float* __restrict__ shared_out, float* __restrict__ species_out,
    float* __restrict__ cond) {
    int e = blockIdx.x * blockDim.x + threadIdx.x;
    if (e >= NEDGES) return;
    float x[16];
    #pragma unroll
    for (int k = 0; k < 16; ++k) x[k] = ef[e * 16 + k];
    float s = mlp16_32_32_1(x, sW1, sb1, sW2, sb2, sW3, sb3);
    float p = mlp16_32_32_1(x, pW1, pb1, pW2, pb2, pW3, pb3);
    shared_out[e]  = s;
    species_out[e] = p;
    float z = s + p;
    float sp = fmaxf(z, 0.0f) + log1pf(expf(-fabsf(z)));   // stable softplus
    cond[e] = 1.0f / (sp + 1.0e-4f);
}

// ---------------------------------------------------------------------------
// M = L + (1/n) * ones  (init to 1/n, scatter Laplacian with atomics)
// ---------------------------------------------------------------------------
__global__ __launch_bounds__(256) void init_M_kernel(float* __restrict__ M) {
    size_t i = (size_t)blockIdx.x * blockDim.x + threadIdx.x;
    if (i < (size_t)MAT_ELEMS) M[i] = 1.0f / (float)NNODES;
}

__global__ __launch_bounds__(256) void scatter_kernel(const int* __restrict__ ei,
                                                      const float* __restrict__ cond,
                                                      float* __restrict__ M) {
    int e = blockIdx.x * blockDim.x + threadIdx.x;
    if (e >= NEDGES) return;
    int i = ei[2 * e + 0];
    int j = ei[2 * e + 1];
    float c = cond[e];
    atomicAdd(&M[(size_t)i * NNODES + i],  c);
    atomicAdd(&M[(size_t)j * NNODES + j],  c);
    atomicAdd(&M[(size_t)i * NNODES + j], -c);
    atomicAdd(&M[(size_t)j * NNODES + i], -c);
}

// ---------------------------------------------------------------------------
// Newton-Schulz seed: c = 1 / (||M||_1 * ||M||_inf) = 1 / s^2 (M symmetric)
// ---------------------------------------------------------------------------
__global__ __launch_bounds__(256) void row_abs_sum_kernel(const float* __restrict__ M,
                                                          float* __restrict__ rowsum) {
    __shared__ float red[256];
    int row = blockIdx.x;
    float s = 0.0f;
    for (int c = threadIdx.x; c < NNODES; c += 256) s += fabsf(M[(size_t)row * NNODES + c]);
    red[threadIdx.x] = s;
    __syncthreads();
    for (int w = 128; w > 0; w >>= 1) {
        if (threadIdx.x < (unsigned)w) red[threadIdx.x] += red[threadIdx.x + w];
        __syncthreads();
    }
    if (threadIdx.x == 0) rowsum[row] = red[0];
}

__global__ __launch_bounds__(256) void max_inv_sq_kernel(const float* __restrict__ rowsum,
                                                         float* __restrict__ cscale) {
    __shared__ float red[256];
    float m = 0.0f;
    for (int i = threadIdx.x; i < NNODES; i += 256) m = fmaxf(m, rowsum[i]);
    red[threadIdx.x] = m;
    __syncthreads();
    for (int w = 128; w > 0; w >>= 1) {
        if (threadIdx.x < (unsigned)w) red[threadIdx.x] = fmaxf(red[threadIdx.x], red[threadIdx.x + w]);
        __syncthreads();
    }
    if (threadIdx.x == 0) cscale[0] = 1.0f / (red[0] * red[0]);
}

__global__ __launch_bounds__(256) void scale_init_kernel(const float* __restrict__ M,
                                                         const float* __restrict__ cscale,
                                                         float* __restrict__ X) {
    size_t i = (size_t)blockIdx.x * blockDim.x + threadIdx.x;
    if (i < (size_t)MAT_ELEMS) X[i] = cscale[0] * M[i];
}

// ---------------------------------------------------------------------------
// WMMA f32 GEMM: C = A*B (4096^3, NN, row-major). If Xepi != null, fused
// epilogue C = 2*Xepi - A*B (the Newton-Schulz update X(2I - MX)).
// Block: 256 threads = 8 waves. Block tile 128x128, K-step 16, LDS double-
// buffered with register staging: one barrier per K-step, next-tile global
// loads overlap the 32-WMMA compute block. Wave tile 32x64 = 2x4 accums.
// ---------------------------------------------------------------------------
__device__ __forceinline__ void load_tiles(const float* __restrict__ A,
                                           const float* __restrict__ B,
                                           int bm, int bn, int kt, int t,
                                           v4f a_reg[2], v4f b_reg[2]) {
    #pragma unroll
    for (int i = 0; i < 2; ++i) {           // A tile 128x16 = 512 float4
        int id4 = t + i * 256;
        int row = id4 >> 2;                 // 4 float4 per 16-wide row
        int c4  = (id4 & 3) << 2;
        a_reg[i] = *(const v4f*)(A + (size_t)(bm + row) * NNODES + kt + c4);
    }
    #pragma unroll
    for (int i = 0; i < 2; ++i) {           // B tile 16x128 = 512 float4
        int id4 = t + i * 256;
        int row = id4 >> 5;                 // 32 float4 per 128-wide row
        int c4  = (id4 & 31) << 2;
        b_reg[i] = *(const v4f*)(B + (size_t)(kt + row) * NNODES + bn + c4);
    }
}

__device__ __forceinline__ void store_tiles(float* __restrict__ As_,
                                            float* __restrict__ Bs_,
                                            int t, const v4f a_reg[2], const v4f b_reg[2]) {
    #pragma unroll
    for (int i = 0; i < 2; ++i) {
        int id4 = t + i * 256;
        int row = id4 >> 2;
        int c4  = (id4 & 3) << 2;
        *(v4f*)(&As_[row * LDA_S + c4]) = a_reg[i];
    }
    #pragma unroll
    for (int i = 0; i < 2; ++i) {
        int id4 = t + i * 256;
        int row = id4 >> 5;
        int c4  = (id4 & 31) << 2;
        *(v4f*)(&Bs_[row * LDB_S + c4]) = b_reg[i];
    }
}

__global__ __launch_bounds__(256) void wmma_gemm_nn(const float* __restrict__ A,
                                                    const float* __restrict__ B,
                                                    float* __restrict__ C,
                                                    const float* __restrict__ Xepi) {
    __shared__ float As[2][BM * LDA_S];
    __shared__ float Bs[2][BK * LDB_S];

    const int t    = threadIdx.x;
    const int lane = t & 31;
    const int wave = t >> 5;
    const int bm   = blockIdx.y * BM;
    const int bn   = blockIdx.x * BN;
    const int wm   = (wave & 3) * 32;    // 4 waves tile M: 0,32,64,96
    const int wn   = (wave >> 2) * 64;   // 2 waves tile N: 0,64
    const int l15  = lane & 15;
    const int kb   = (lane >> 4) * 2;    // lanes 16-31 hold K+2/K+3 halves

    v8f acc[2][4] = {};
    v4f a_reg[2], b_reg[2];

    // Prologue: stage tile kt=0
    load_tiles(A, B, bm, bn, 0, t, a_reg, b_reg);
    store_tiles(As[0], Bs[0], t, a_reg, b_reg);
    __syncthreads();

    int cur = 0;
    for (int kt = 0; kt < NNODES; kt += BK) {
        const bool has_next = (kt + BK) < NNODES;
        if (has_next) load_tiles(A, B, bm, bn, kt + BK, t, a_reg, b_reg);

        const float* __restrict__ Asc = As[cur];
        const float* __restrict__ Bsc = Bs[cur];
        #pragma unroll
        for (int kk = 0; kk < BK; kk += 4) {
            v2f a[2], b[4];
            // A frag 16x4: lanes 0-15 -> (K,K+1), lanes 16-31 -> (K+2,K+3)
            #pragma unroll
            for (int mi = 0; mi < 2; ++mi) {
                a[mi].x = Asc[(wm + mi * 16 + l15) * LDA_S + kk + kb];
                a[mi].y = Asc[(wm + mi * 16 + l15) * LDA_S + kk + kb + 1];
            }
            // B frag 4x16: mirrored K split across lane halves
            #pragma unroll
            for (int ni = 0; ni < 4; ++ni) {
                b[ni].x = Bsc[(kk + kb)     * LDB_S + wn + ni * 16 + l15];
                b[ni].y = Bsc[(kk + kb + 1) * LDB_S + wn + ni * 16 + l15];
            }
            #pragma unroll
            for (int mi = 0; mi < 2; ++mi)
                #pragma unroll
                for (int ni = 0; ni < 4; ++ni)
                    acc[mi][ni] = __builtin_amdgcn_wmma_f32_16x16x4_f32(
                        false, a[mi], false, b[ni], (short)0, acc[mi][ni], false, false);
        }

        if (has_next) store_tiles(As[cur ^ 1], Bs[cur ^ 1], t, a_reg, b_reg);
        __syncthreads();
        cur ^= 1;
    }

    // Store: C/D layout -> VGPR r holds M=r (lanes 0-15) / M=r+8 (lanes 16-31)
    const int rbase = (lane >> 4) * 8;
    #pragma unroll
    for (int mi = 0; mi < 2; ++mi) {
        #pragma unroll
        for (int ni = 0; ni < 4; ++ni) {
            #pragma unroll
            for (int r = 0; r < 8; ++r) {
                size_t gm  = (size_t)(bm + wm + mi * 16 + rbase + r);
                size_t gn  = (size_t)(bn + wn + ni * 16 + l15);
                size_t off = gm * NNODES + gn;
                float v = acc[mi][ni][r];
                if (Xepi) v = 2.0f * Xepi[off] - v;
                C[off] = v;
            }
        }
    }
}

// ---------------------------------------------------------------------------
// R_ij = Minv_ii + Minv_jj - 2*Minv_ij  (J/n terms of pinv cancel exactly)
// ---------------------------------------------------------------------------
__global__ __launch_bounds__(256) void diag_kernel(const float* __restrict__ X,
                                                   float* __restrict__ diag) {
    int i = blockIdx.x * blockDim.x + threadIdx.x;
    if (i < NNODES) diag[i] = X[(size_t)i * NNODES + i];
}

__global__ __launch_bounds__(256) void resist_kernel(const float* __restrict__ X,
                                                     const float* __restrict__ diag,
                                                     float* __restrict__ R) {
    size_t idx = (size_t)blockIdx.x * blockDim.x + threadIdx.x;
    if (idx >= (size_t)MAT_ELEMS) return;
    int i = (int)(idx / NNODES);
    int j = (int)(idx % NNODES);
    R[idx] = diag[i] + diag[j] - 2.0f * X[idx];
}

// ---------------------------------------------------------------------------
extern "C" void kernel_launch(void* const* d_in, const int* in_sizes, int n_in,
                              void* d_out, int out_size, void* d_ws, size_t ws_size,
                              hipStream_t stream) {
    const int*   edge_index = (const int*)d_in[1];
    const float* edge_feat  = (const float*)d_in[2];
    const float* sW1 = (const float*)d_in[4];
    const float* sb1 = (const float*)d_in[5];
    const float* sW2 = (const float*)d_in[6];
    const float* sb2 = (const float*)d_in[7];
    const float* sW3 = (const float*)d_in[8];
    const float* sb3 = (const float*)d_in[9];
    const float* pW1 = (const float*)d_in[10];
    const float* pb1 = (const float*)d_in[11];
    const float* pW2 = (const float*)d_in[12];
    const float* pb2 = (const float*)d_in[13];
    const float* pW3 = (const float*)d_in[14];
    const float* pb3 = (const float*)d_in[15];

    float* out         = (float*)d_out;
    float* R_out       = out;                       // 4096*4096
    float* shared_out  = out + (size_t)MAT_ELEMS;   // 131072
    float* species_out = shared_out + NEDGES;       // 131072

    // Workspace layout (floats)
    float* ws     = (float*)d_ws;
    float* cond   = ws;                    // 131072
    float* rowsum = ws + NEDGES;           // 4096
    float* cscale = rowsum + NNODES;       // 1
    float* diag   = cscale + 64;           // 4096
    float* M      = ws + (1 << 18);        // 4096^2  (offset 1 MiB, clear of the small buffers)
    float* T      = M  + (size_t)MAT_ELEMS;
    float* Xa     = T  + (size_t)MAT_ELEMS;
    float* Xb     = R_out;                 // R region of d_out doubles as ping-pong buffer

    // 1) Edge MLPs + conductance
    mlp_edges_kernel<<<NEDGES / 256, 256, 0, stream>>>(
        edge_feat, sW1, sb1, sW2, sb2, sW3, sb3,
        pW1, pb1, pW2, pb2, pW3, pb3,
        shared_out, species_out, cond);

    // 2) M = L + J/n
    init_M_kernel<<<MAT_ELEMS / 256, 256, 0, stream>>>(M);
    scatter_kernel<<<NEDGES / 256, 256, 0, stream>>>(edge_index, cond, M);

    // 3) Newton-Schulz seed X0 = M / (||M||_1 ||M||_inf)
    row_abs_sum_kernel<<<NNODES, 256, 0, stream>>>(M, rowsum);
    max_inv_sq_kernel<<<1, 256, 0, stream>>>(rowsum, cscale);
    scale_init_kernel<<<MAT_ELEMS / 256, 256, 0, stream>>>(M, cscale, Xa);

    // 4) X <- X(2I - MX): T = M*X ; Xnext = 2*X - X*T   (WMMA f32 GEMMs)
    dim3 ggrid(NNODES / BN, NNODES / BM);
    float* Xcur = Xa;
    float* Xnxt = Xb;
    for (int it = 0; it < NS_ITERS; ++it) {
        wmma_gemm_nn<<<ggrid, 256, 0, stream>>>(M, Xcur, T, nullptr);
        wmma_gemm_nn<<<ggrid, 256, 0, stream>>>(Xcur, T, Xnxt, Xcur);
        float* tmp = Xcur; Xcur = Xnxt; Xnxt = tmp;
    }
    // NS_ITERS is even -> Xcur == Xa here; d_out R region is free again.

    // 5) R from inv(M) (J/n cancels in the resistance formula)
    diag_kernel<<<NNODES / 256, 256, 0, stream>>>(Xcur, diag);
    resist_kernel<<<MAT_ELEMS / 256, 256, 0, stream>>>(Xcur, diag, R_out);
}